// Multimodal_LDM_77807627534714
// MI455X (gfx1250) — compile-verified
//
#include <hip/hip_runtime.h>
#include <math.h>

#define N_PP 100000
#define N_A  50000
#define N_AP (N_PP + N_A)
#define DIM  16
#define BSET 8192
#define NEDGE 500000

#define EDGE_BLOCKS 1024
#define NL_GRID_X   8             // tj segments per tile row
#define NL_GRID_Y   (BSET / 16)   // 512 tile rows
#define NL_PARTS    (NL_GRID_X * NL_GRID_Y)  // 4096 partials

typedef __attribute__((ext_vector_type(2))) float v2f;
typedef __attribute__((ext_vector_type(8))) float v8f;

// ---- workspace layout (float element offsets) ----
constexpr int OFF_XS   = 0;                     // p_star gathered  [BSET*DIM]
constexpr int OFF_YP   = OFF_XS  + BSET * DIM;  // p gathered       [BSET*DIM]
constexpr int OFF_YA   = OFF_YP  + BSET * DIM;  // a gathered       [BSET*DIM]
constexpr int OFF_XX   = OFF_YA  + BSET * DIM;  // |x|^2            [BSET]
constexpr int OFF_YYP  = OFF_XX  + BSET;
constexpr int OFF_YYA  = OFF_YYP + BSET;
constexpr int OFF_GI   = OFF_YYA + BSET;        // gamma[p_star_nodes]
constexpr int OFF_GJP  = OFF_GI  + BSET;        // gamma[p_nodes+N_PP]
constexpr int OFF_BI   = OFF_GJP + BSET;        // beta[p_star_nodes]
constexpr int OFF_BJA  = OFF_BI  + BSET;        // beta[a_nodes]
constexpr int OFF_PLPP = OFF_BJA + BSET;        // link partials pp [EDGE_BLOCKS]
constexpr int OFF_PLAP = OFF_PLPP + EDGE_BLOCKS;
constexpr int OFF_PNPP = OFF_PLAP + EDGE_BLOCKS; // nonlink partials pp [NL_PARTS]
constexpr int OFF_PNAP = OFF_PNPP + NL_PARTS;

#define L2E 1.4426950408889634f   // log2(e)

// ------------------------------------------------------------------
// Gather sampled node features + squared norms + rate vectors into ws
// ------------------------------------------------------------------
__global__ void gather_kernel(const float* __restrict__ p,
                              const float* __restrict__ p_star,
                              const float* __restrict__ a,
                              const float* __restrict__ beta,
                              const float* __restrict__ gamma,
                              const int* __restrict__ psn,
                              const int* __restrict__ pn,
                              const int* __restrict__ an,
                              float* __restrict__ ws) {
  int i = blockIdx.x * blockDim.x + threadIdx.x;
  if (i >= BSET) return;

  int s = psn[i];
  float acc = 0.f;
#pragma unroll
  for (int d = 0; d < DIM; ++d) {
    float v = p_star[(size_t)s * DIM + d];
    ws[OFF_XS + i * DIM + d] = v;
    acc += v * v;
  }
  ws[OFF_XX + i] = acc;
  ws[OFF_GI + i] = gamma[s];
  ws[OFF_BI + i] = beta[s];

  int q = pn[i];
  acc = 0.f;
#pragma unroll
  for (int d = 0; d < DIM; ++d) {
    float v = p[(size_t)q * DIM + d];
    ws[OFF_YP + i * DIM + d] = v;
    acc += v * v;
  }
  ws[OFF_YYP + i] = acc;
  ws[OFF_GJP + i] = gamma[q + N_PP];

  int t = an[i];
  acc = 0.f;
#pragma unroll
  for (int d = 0; d < DIM; ++d) {
    float v = a[(size_t)(t - N_PP) * DIM + d];
    ws[OFF_YA + i * DIM + d] = v;
    acc += v * v;
  }
  ws[OFF_YYA + i] = acc;
  ws[OFF_BJA + i] = beta[t];
}

// ------------------------------------------------------------------
// Link term: sum over edges of W[e0] + W[e1+wBoff] - pdist
// ------------------------------------------------------------------
__global__ void edge_kernel(const int* __restrict__ e0,
                            const int* __restrict__ e1,
                            const float* __restrict__ X,
                            const float* __restrict__ Y,
                            const float* __restrict__ W,
                            int wBoff, int yoff,
                            float* __restrict__ partial) {
  float acc = 0.f;
  for (int k = blockIdx.x * blockDim.x + threadIdx.x; k < NEDGE;
       k += gridDim.x * blockDim.x) {
    int i = e0[k];
    int j = e1[k];
    const float* xr = X + (size_t)i * DIM;
    const float* yr = Y + (size_t)(j - yoff) * DIM;
    float s = 1e-12f;
#pragma unroll
    for (int d = 0; d < DIM; ++d) {
      float df = xr[d] - yr[d];
      s += df * df;
    }
    // s >= 1e-12 (normal range): raw v_sqrt_f32 is safe and ~1 ulp
    acc += W[i] + W[j + wBoff] - __builtin_amdgcn_sqrtf(s);
  }
  __shared__ float red[256];
  red[threadIdx.x] = acc;
  __syncthreads();
  for (int s2 = 128; s2 > 0; s2 >>= 1) {
    if (threadIdx.x < (unsigned)s2) red[threadIdx.x] += red[threadIdx.x + s2];
    __syncthreads();
  }
  if (threadIdx.x == 0) partial[blockIdx.x] = red[0];
}

// ------------------------------------------------------------------
// Non-link term. blockIdx.y = tile row (ti). Each wave sweeps tiles
// tj = blockIdx.x*8 + wave, step NL_GRID_X*8. The A fragment, |x|^2
// and gI*log2e for the 16 rows are register-resident; the inner loop
// is B-load + 4x V_WMMA_F32_16X16X4_F32 + native sqrt/exp2 epilogue.
// ------------------------------------------------------------------
__global__ void nonlink_kernel(const float* __restrict__ X,
                               const float* __restrict__ Y,
                               const float* __restrict__ xx,
                               const float* __restrict__ yy,
                               const float* __restrict__ gI,
                               const float* __restrict__ gJ,
                               int triu,
                               float* __restrict__ partial) {
  constexpr int TPR = BSET / 16;      // 512 tiles per row

  const int lane = threadIdx.x & 31;
  const int wave = threadIdx.x >> 5;
  const int wavesPerBlk = blockDim.x >> 5;      // 8
  const int r  = lane & 15;           // row of A / row of B / column N
  const int hi = lane >> 4;           // lane-half selector

  const int ti = blockIdx.y;
  const int i0 = ti * 16;
  const int tjStart  = blockIdx.x * wavesPerBlk + wave;
  const int tjStride = gridDim.x * wavesPerBlk;

  // ---- hoisted per-row state (constant over the tj sweep) ----
  const float* xr = X + (size_t)(i0 + r) * DIM;
  float av0[8];                        // A fragment: K = 4kc + 2*hi + {0,1}
#pragma unroll
  for (int kc = 0; kc < 4; ++kc) {
    av0[2 * kc]     = xr[4 * kc + 2 * hi];
    av0[2 * kc + 1] = xr[4 * kc + 2 * hi + 1];
  }
  float xi2[8], giL[8];
#pragma unroll
  for (int v = 0; v < 8; ++v) {
    int ig = i0 + v + 8 * hi;
    xi2[v] = xx[ig];
    giL[v] = gI[ig] * L2E;             // pre-scaled by log2(e)
  }

  float acc = 0.f;
  for (int tj = tjStart; tj < TPR; tj += tjStride) {
    if (triu && tj < ti) continue;    // wave-uniform: tile fully masked
    int j0 = tj * 16;

    const float* yr = Y + (size_t)(j0 + r) * DIM;
    v8f c = {};
#pragma unroll
    for (int kc = 0; kc < 4; ++kc) {
      v2f av = { av0[2 * kc], av0[2 * kc + 1] };
      v2f bv = { yr[4 * kc + 2 * hi], yr[4 * kc + 2 * hi + 1] };
      c = __builtin_amdgcn_wmma_f32_16x16x4_f32(
          /*neg_a=*/false, av, /*neg_b=*/false, bv,
          /*c_mod=*/(short)0, c, /*reuse_a=*/false, /*reuse_b=*/false);
    }

    // C/D layout: N = lane&15; M = vgpr + 8*(lane>>4)
    int jg = j0 + r;
    float yyj = yy[jg];
    float gjL = gJ[jg] * L2E;

    // per-tile hoists: s8 = |x|^2 + |y|^2 ; e8 = (gi+gj)*log2e
    float s8[8], e8[8];
#pragma unroll
    for (int v = 0; v < 8; ++v) {
      s8[v] = xi2[v] + yyj;
      e8[v] = giL[v] + gjL;
    }

    if (!triu || tj > ti) {
      // fully unmasked tile
#pragma unroll
      for (int v = 0; v < 8; ++v) {
        float sq   = __builtin_fmaf(c[v], -2.0f, s8[v]);
        float dist = __builtin_amdgcn_sqrtf(fmaxf(sq, 1e-12f));
        acc += __builtin_amdgcn_exp2f(__builtin_fmaf(dist, -L2E, e8[v]));
      }
    } else {
      // diagonal tile: per-element triu(k=1) mask
#pragma unroll
      for (int v = 0; v < 8; ++v) {
        int ig = i0 + v + 8 * hi;
        float sq   = __builtin_fmaf(c[v], -2.0f, s8[v]);
        float dist = __builtin_amdgcn_sqrtf(fmaxf(sq, 1e-12f));
        float val  = __builtin_amdgcn_exp2f(__builtin_fmaf(dist, -L2E, e8[v]));
        acc += (jg > ig) ? val : 0.f;
      }
    }
  }

  // wave32 butterfly reduce
#pragma unroll
  for (int off = 16; off > 0; off >>= 1) acc += __shfl_xor(acc, off, 32);

  __shared__ float wsum[32];
  if (lane == 0) wsum[wave] = acc;
  __syncthreads();
  if (threadIdx.x == 0) {
    float s = 0.f;
    for (int w = 0; w < wavesPerBlk; ++w) s += wsum[w];
    partial[blockIdx.y * gridDim.x + blockIdx.x] = s;
  }
}

// ------------------------------------------------------------------
// Deterministic final combine
// ------------------------------------------------------------------
__global__ void finalize_kernel(const float* __restrict__ ws,
                                float* __restrict__ out) {
  float lpp = 0.f, lap = 0.f, npp = 0.f, nap = 0.f;
  for (int i = 0; i < EDGE_BLOCKS; ++i) lpp += ws[OFF_PLPP + i];
  for (int i = 0; i < EDGE_BLOCKS; ++i) lap += ws[OFF_PLAP + i];
  for (int i = 0; i < NL_PARTS; ++i)    npp += ws[OFF_PNPP + i];
  for (int i = 0; i < NL_PARTS; ++i)    nap += ws[OFF_PNAP + i];
  float nll_pp = -(lpp - npp);
  float nll_ap = -(lap - nap);
  out[0] = 0.5f * nll_pp / (float)BSET + 0.5f * nll_ap / (float)BSET;
}

extern "C" void kernel_launch(void* const* d_in, const int* in_sizes, int n_in,
                              void* d_out, int out_size, void* d_ws, size_t ws_size,
                              hipStream_t stream) {
  const float* p       = (const float*)d_in[0];
  const float* p_star  = (const float*)d_in[1];
  const float* a       = (const float*)d_in[2];
  const float* beta    = (const float*)d_in[3];
  const float* gamma   = (const float*)d_in[4];
  const int* edges_pp  = (const int*)d_in[5];
  const int* edges_ap  = (const int*)d_in[6];
  const int* psn       = (const int*)d_in[7];
  const int* pn        = (const int*)d_in[8];
  const int* an        = (const int*)d_in[9];
  float* ws  = (float*)d_ws;
  float* out = (float*)d_out;

  gather_kernel<<<BSET / 256, 256, 0, stream>>>(p, p_star, a, beta, gamma,
                                                psn, pn, an, ws);

  // pp link: gamma[e0] + gamma[e1+N_PP] - dist(p_star[e0], p[e1])
  edge_kernel<<<EDGE_BLOCKS, 256, 0, stream>>>(edges_pp, edges_pp + NEDGE,
                                               p_star, p, gamma, N_PP, 0,
                                               ws + OFF_PLPP);
  // ap link: beta[e0] + beta[e1] - dist(p_star[e0], a[e1-N_PP])
  edge_kernel<<<EDGE_BLOCKS, 256, 0, stream>>>(edges_ap, edges_ap + NEDGE,
                                               p_star, a, beta, 0, N_PP,
                                               ws + OFF_PLAP);

  dim3 nlGrid(NL_GRID_X, NL_GRID_Y);
  nonlink_kernel<<<nlGrid, 256, 0, stream>>>(ws + OFF_XS, ws + OFF_YP,
                                             ws + OFF_XX, ws + OFF_YYP,
                                             ws + OFF_GI, ws + OFF_GJP,
                                             /*triu=*/1, ws + OFF_PNPP);
  nonlink_kernel<<<nlGrid, 256, 0, stream>>>(ws + OFF_XS, ws + OFF_YA,
                                             ws + OFF_XX, ws + OFF_YYA,
                                             ws + OFF_BI, ws + OFF_BJA,
                                             /*triu=*/0, ws + OFF_PNAP);

  finalize_kernel<<<1, 1, 0, stream>>>(ws, out);
}